// c_7_75445395521979
// MI455X (gfx1250) — compile-verified
//
#include <hip/hip_runtime.h>

// ---------------------------------------------------------------------------
// CDNA5 async copy helpers (GLOBAL_LOAD_ASYNC_TO_LDS_*, tracked by ASYNCcnt).
// Operands per ISA 10.x/15.18: VDST = LDS byte address VGPR, VADDR = 32-bit
// per-lane byte offset, SADDR = 64-bit uniform base (GVS mode).
// ---------------------------------------------------------------------------
__device__ __forceinline__ void async_copy_b128(unsigned lds_byte_off,
                                                unsigned glb_byte_off,
                                                const float* base) {
    asm volatile("global_load_async_to_lds_b128 %0, %1, %2"
                 :: "v"(lds_byte_off), "v"(glb_byte_off), "s"(base)
                 : "memory");
}
__device__ __forceinline__ void async_wait0() {
    asm volatile("s_wait_asynccnt 0" ::: "memory");
}

// ---------------------------------------------------------------------------
// Stage 1: conv1 (3->3, 3x3, stride2) + maxpool 3x3/3 + relu
// nodes viewed as [1280][3][224][224] -> pool1 [1280][3][37][37] (stride 4112)
// Memory-bound: 770MB @ 23.3TB/s ~= 33us floor. Each block handles TWO pooled
// rows (13-row band, 1.08x read amplification) with double-buffered
// async-to-LDS staging: band1's loads are in flight while band0 computes.
// ---------------------------------------------------------------------------
__device__ __forceinline__ void conv1_issue_band(const float* src, unsigned lds_base,
                                                 int tid) {
    // 3 channels x 7 rows x 224 cols = 1176 float4 = 18816 B
    for (int i = tid; i < 1176; i += 128) {
        const int ci = i / 392;                  // 392 = 7*224/4
        const int r  = (i % 392) / 56;           // 56  = 224/4
        const int c4 = i % 56;
        const unsigned goff = (unsigned)(ci * 50176 + r * 224 + c4 * 4) * 4u;
        async_copy_b128(lds_base + (unsigned)i * 16u, goff, src);
    }
}

__device__ __forceinline__ void conv1_compute(const float* t, int img, int po,
                                              int tid,
                                              const float* __restrict__ w1g,
                                              const float* __restrict__ b1g,
                                              float* __restrict__ pool1) {
    if (tid < 111) {                             // 3 out-channels x 37 pooled cols
        const int oc = tid / 37;
        const int qo = tid % 37;
        float w[27];
#pragma unroll
        for (int j = 0; j < 27; ++j) w[j] = w1g[oc * 27 + j];
        const float bias = b1g[oc];

        float m = -3.4e38f;
#pragma unroll
        for (int pr = 0; pr < 3; ++pr) {
#pragma unroll
            for (int pc = 0; pc < 3; ++pc) {
                float acc = bias;
                const int cbase = 6 * qo + 2 * pc;
#pragma unroll
                for (int ci = 0; ci < 3; ++ci)
#pragma unroll
                    for (int kh = 0; kh < 3; ++kh)
#pragma unroll
                        for (int kw = 0; kw < 3; ++kw)
                            acc += t[ci * 1568 + (2 * pr + kh) * 224 + cbase + kw]
                                   * w[(ci * 3 + kh) * 3 + kw];
                m = fmaxf(m, acc);
            }
        }
        pool1[(size_t)img * 4112 + oc * 1369 + po * 37 + qo] = fmaxf(m, 0.0f);
    }
}

__global__ __launch_bounds__(128) void k_conv1(const float* __restrict__ nodes,
                                               const float* __restrict__ w1g,
                                               const float* __restrict__ b1g,
                                               float* __restrict__ pool1) {
    __shared__ float tile[2 * 4704];             // double buffer, 37.6 KB
    const int img = blockIdx.x;
    const int po0 = blockIdx.y * 2;              // pooled rows po0, po0+1
    const int tid = threadIdx.x;
    const bool has2 = (po0 + 1) < 37;            // block-uniform

    const float* src0 = nodes + (size_t)img * 3 * 224 * 224 + (size_t)(6 * po0) * 224;
    const float* src1 = src0 + 6 * 224;

    conv1_issue_band(src0, 0u, tid);             // band 0 -> buffer 0
    async_wait0();
    __syncthreads();

    if (has2) conv1_issue_band(src1, 18816u, tid);   // band 1 in flight

    conv1_compute(tile, img, po0, tid, w1g, b1g, pool1);

    if (has2) {
        async_wait0();
        __syncthreads();
        conv1_compute(tile + 4704, img, po0 + 1, tid, w1g, b1g, pool1);
    }
}

// ---------------------------------------------------------------------------
// Stage 2: conv2 (3->1, 3x3, stride2) + maxpool 3x3/3 + relu + lin (36->6)
// pool1 [1280][4112] (21MB, L2-resident) -> nodes_feature nf [1280][6]
// Async-to-LDS staging; per-image stride padded to 4112 floats so b128 copies
// stay 16B-aligned (1027 float4 covers the 4107 live floats).
// ---------------------------------------------------------------------------
__global__ __launch_bounds__(128) void k_conv2(const float* __restrict__ pool1,
                                               const float* __restrict__ w2g,
                                               const float* __restrict__ b2g,
                                               const float* __restrict__ lin_w,
                                               const float* __restrict__ lin_b,
                                               float* __restrict__ nf) {
    __shared__ float tile[1027 * 4];             // 4108 floats (4107 live)
    __shared__ float f36[36];
    const int img = blockIdx.x;
    const int tid = threadIdx.x;

    const float* src = pool1 + (size_t)img * 4112;
    for (int i = tid; i < 1027; i += 128)
        async_copy_b128((unsigned)i * 16u, (unsigned)i * 16u, src);
    async_wait0();
    __syncthreads();

    if (tid < 36) {
        float w[27];
#pragma unroll
        for (int j = 0; j < 27; ++j) w[j] = w2g[j];
        const float bias = b2g[0];
        const int pp = tid / 6, qq = tid % 6;
        float m = -3.4e38f;
#pragma unroll
        for (int pr = 0; pr < 3; ++pr) {
#pragma unroll
            for (int pc = 0; pc < 3; ++pc) {
                const int r = 3 * pp + pr;       // conv row 0..17
                const int c = 3 * qq + pc;
                float acc = bias;
#pragma unroll
                for (int ci = 0; ci < 3; ++ci)
#pragma unroll
                    for (int kh = 0; kh < 3; ++kh)
#pragma unroll
                        for (int kw = 0; kw < 3; ++kw)
                            acc += tile[ci * 1369 + (2 * r + kh) * 37 + (2 * c + kw)]
                                   * w[(ci * 3 + kh) * 3 + kw];
                m = fmaxf(m, acc);
            }
        }
        f36[tid] = fmaxf(m, 0.0f);
    }
    __syncthreads();

    if (tid < 6) {
        float acc = lin_b[tid];
#pragma unroll
        for (int i = 0; i < 36; ++i) acc += f36[i] * lin_w[tid * 36 + i];
        nf[(size_t)img * 6 + tid] = acc;
    }
}

// ---------------------------------------------------------------------------
// Stage 3: message passing + concat -> feat [64][240]
// ---------------------------------------------------------------------------
__global__ __launch_bounds__(128) void k_graph(const float* __restrict__ pos,
                                               const float* __restrict__ attmap,
                                               const float* __restrict__ frame_w,
                                               const float* __restrict__ frame_b,
                                               const float* __restrict__ last_w,
                                               const float* __restrict__ last_b,
                                               const float* __restrict__ nf,
                                               float* __restrict__ feat) {
    __shared__ float msg[120];                   // [f][j][d]
    const int b = blockIdx.x;
    const int tid = threadIdx.x;

    if (tid < 120) {
        const int f = tid / 24, j = (tid / 6) % 4, d = tid % 6;
        const float* pp = pos + ((size_t)(b * 5 + f) * 4 + j) * 6;
        float acc = frame_b[d];
#pragma unroll
        for (int k = 0; k < 6; ++k) acc += pp[k] * frame_w[d * 6 + k];
        msg[tid] = acc;
    }
    __syncthreads();

    if (tid < 120) {
        const int f = tid / 24, n = (tid / 6) % 4, d = tid % 6;
        float acc = 0.0f;
#pragma unroll
        for (int j = 0; j < 4; ++j)
            acc += attmap[((size_t)(b * 5 + f) * 4 + j) * 4 + n] * msg[(f * 4 + j) * 6 + d];
        if (f >= 1) {                            // + last(pos[:, f-1])
            const float* pp = pos + ((size_t)(b * 5 + (f - 1)) * 4 + n) * 6;
            float l = last_b[d];
#pragma unroll
            for (int k = 0; k < 6; ++k) l += pp[k] * last_w[d * 6 + k];
            acc += l;
        }
        const int base = b * 240 + (f * 4 + n) * 12;
        feat[base + d]     = nf[((size_t)(b * 20 + f * 4 + n)) * 6 + d];
        feat[base + 6 + d] = acc;
    }
}

// ---------------------------------------------------------------------------
// Stage 4: FC stack with fp32 WMMA (V_WMMA_F32_16X16X4_F32).
//   fc1: [64,240]x[240,120]^T +relu   fc2: [64,120]x[120,60]^T +relu
//   fc3: [64,60]x[60,6]^T -> out [64,6]
// One block, 8 waves; each wave owns 16x16 tiles. EXEC kept all-1s around
// every WMMA: OOB weight columns use clamped indices + multiplicative mask.
// Fragment layout (ISA 7.12.2): A/B lanes 0-15 carry K=k,k+1 and lanes 16-31
// carry K=k+2,k+3; C/D VGPR v holds rows m0+v (lanes 0-15) / m0+v+8 (16-31).
// ---------------------------------------------------------------------------
typedef float v2f __attribute__((ext_vector_type(2)));
typedef float v8f __attribute__((ext_vector_type(8)));

__global__ __launch_bounds__(256) void k_fc(const float* __restrict__ feat,
                                            const float* __restrict__ fc1_w,
                                            const float* __restrict__ fc1_b,
                                            const float* __restrict__ fc2_w,
                                            const float* __restrict__ fc2_b,
                                            const float* __restrict__ fc3_w,
                                            const float* __restrict__ fc3_b,
                                            float* __restrict__ out) {
    __shared__ float h1[64 * 120];               // 30 KB
    __shared__ float h2[64 * 60];                // 15 KB
    const int tid  = threadIdx.x;
    const int lane = tid & 31;
    const int wave = tid >> 5;                   // 0..7
    const int hl   = lane >> 4;                  // 0/1: which K-pair this lane holds
    const int l16  = lane & 15;

    // ---- fc1 ---- 4 M-tiles x 8 N-tiles (cols padded 120->128)
    for (int t = wave; t < 32; t += 8) {
        const int m0 = (t & 3) * 16;
        const int n0 = (t >> 2) * 16;
        const int arow  = m0 + l16;
        const int bcol  = n0 + l16;
        const int bcolc = bcol < 120 ? bcol : 119;
        const float bm  = bcol < 120 ? 1.0f : 0.0f;
        v8f c = {0.f, 0.f, 0.f, 0.f, 0.f, 0.f, 0.f, 0.f};
        for (int k = 0; k < 240; k += 4) {
            const int kk = k + hl * 2;
            v2f a, bfrag;
            a.x = feat[arow * 240 + kk];
            a.y = feat[arow * 240 + kk + 1];
            bfrag.x = fc1_w[bcolc * 240 + kk] * bm;
            bfrag.y = fc1_w[bcolc * 240 + kk + 1] * bm;
            c = __builtin_amdgcn_wmma_f32_16x16x4_f32(false, a, false, bfrag,
                                                      (short)0, c, false, false);
        }
        const float bias = fc1_b[bcolc];
#pragma unroll
        for (int v = 0; v < 8; ++v) {
            const int row = m0 + v + hl * 8;
            if (bcol < 120) h1[row * 120 + bcol] = fmaxf(c[v] + bias, 0.0f);
        }
    }
    __syncthreads();

    // ---- fc2 ---- 4 M-tiles x 4 N-tiles (cols padded 60->64)
    for (int t = wave; t < 16; t += 8) {
        const int m0 = (t & 3) * 16;
        const int n0 = (t >> 2) * 16;
        const int arow  = m0 + l16;
        const int bcol  = n0 + l16;
        const int bcolc = bcol < 60 ? bcol : 59;
        const float bm  = bcol < 60 ? 1.0f : 0.0f;
        v8f c = {0.f, 0.f, 0.f, 0.f, 0.f, 0.f, 0.f, 0.f};
        for (int k = 0; k < 120; k += 4) {
            const int kk = k + hl * 2;
            v2f a, bfrag;
            a.x = h1[arow * 120 + kk];
            a.y = h1[arow * 120 + kk + 1];
            bfrag.x = fc2_w[bcolc * 120 + kk] * bm;
            bfrag.y = fc2_w[bcolc * 120 + kk + 1] * bm;
            c = __builtin_amdgcn_wmma_f32_16x16x4_f32(false, a, false, bfrag,
                                                      (short)0, c, false, false);
        }
        const float bias = fc2_b[bcolc];
#pragma unroll
        for (int v = 0; v < 8; ++v) {
            const int row = m0 + v + hl * 8;
            if (bcol < 60) h2[row * 60 + bcol] = fmaxf(c[v] + bias, 0.0f);
        }
    }
    __syncthreads();

    // ---- fc3 ---- 4 M-tiles x 1 N-tile (cols padded 6->16); waves 4..7 idle
    for (int t = wave; t < 4; t += 8) {
        const int m0 = t * 16;
        const int arow  = m0 + l16;
        const int bcol  = l16;
        const int bcolc = bcol < 6 ? bcol : 5;
        const float bm  = bcol < 6 ? 1.0f : 0.0f;
        v8f c = {0.f, 0.f, 0.f, 0.f, 0.f, 0.f, 0.f, 0.f};
        for (int k = 0; k < 60; k += 4) {
            const int kk = k + hl * 2;
            v2f a, bfrag;
            a.x = h2[arow * 60 + kk];
            a.y = h2[arow * 60 + kk + 1];
            bfrag.x = fc3_w[bcolc * 60 + kk] * bm;
            bfrag.y = fc3_w[bcolc * 60 + kk + 1] * bm;
            c = __builtin_amdgcn_wmma_f32_16x16x4_f32(false, a, false, bfrag,
                                                      (short)0, c, false, false);
        }
        const float bias = fc3_b[bcolc];
#pragma unroll
        for (int v = 0; v < 8; ++v) {
            const int row = m0 + v + hl * 8;
            if (bcol < 6) out[row * 6 + bcol] = c[v] + bias;
        }
    }
}

// ---------------------------------------------------------------------------
extern "C" void kernel_launch(void* const* d_in, const int* in_sizes, int n_in,
                              void* d_out, int out_size, void* d_ws, size_t ws_size,
                              hipStream_t stream) {
    (void)in_sizes; (void)n_in; (void)out_size; (void)ws_size;
    const float* nodes   = (const float*)d_in[0];
    const float* pos     = (const float*)d_in[1];
    const float* attmap  = (const float*)d_in[2];
    // d_in[3] = depths (unused in reference)
    const float* conv1_w = (const float*)d_in[4];
    const float* conv1_b = (const float*)d_in[5];
    const float* conv2_w = (const float*)d_in[6];
    const float* conv2_b = (const float*)d_in[7];
    const float* lin_w   = (const float*)d_in[8];
    const float* lin_b   = (const float*)d_in[9];
    const float* frame_w = (const float*)d_in[10];
    const float* frame_b = (const float*)d_in[11];
    const float* last_w  = (const float*)d_in[12];
    const float* last_b  = (const float*)d_in[13];
    const float* fc1_w   = (const float*)d_in[14];
    const float* fc1_b   = (const float*)d_in[15];
    const float* fc2_w   = (const float*)d_in[16];
    const float* fc2_b   = (const float*)d_in[17];
    const float* fc3_w   = (const float*)d_in[18];
    const float* fc3_b   = (const float*)d_in[19];

    float* ws    = (float*)d_ws;
    float* pool1 = ws;                           // 1280*4112 = 5,263,360 f (padded)
    float* nf    = ws + 5263360;                 // 1280*6    =     7,680 f
    float* feat  = nf + 7680;                    // 64*240    =    15,360 f

    k_conv1<<<dim3(1280, 19), 128, 0, stream>>>(nodes, conv1_w, conv1_b, pool1);
    k_conv2<<<dim3(1280), 128, 0, stream>>>(pool1, conv2_w, conv2_b, lin_w, lin_b, nf);
    k_graph<<<dim3(64), 128, 0, stream>>>(pos, attmap, frame_w, frame_b,
                                          last_w, last_b, nf, feat);
    k_fc<<<dim3(1), 256, 0, stream>>>(feat, fc1_w, fc1_b, fc2_w, fc2_b,
                                      fc3_w, fc3_b, (float*)d_out);
}